// AffineMaskGenerator_88553635709520
// MI455X (gfx1250) — compile-verified
//
#include <hip/hip_runtime.h>

typedef __attribute__((ext_vector_type(2))) float v2f;
typedef __attribute__((ext_vector_type(8))) float v8f;

#define HW_PIX (512 * 512)   // H*W
#define N_B    32
#define N_CLS  8

// ---------------------------------------------------------------------------
// Setup: build P[16][28] in workspace.
//   row k (k<8):  P[k][m] = -1/(2 sigma^2) * sym(i,j) * Q_k[i][j]
//   where Q_k = A_k^T A_k,  A_k = [W_k | -I3 | b_k]  (3x7),
//   (i,j) runs over the upper triangle of 7x7 (28 entries),
//   sym = 1 on diagonal, 2 off-diagonal.
//   rows 8..15 are zero padding (WMMA M=16).
// ---------------------------------------------------------------------------
__global__ void setup_P_kernel(const float* __restrict__ W,
                               const float* __restrict__ bias,
                               const float* __restrict__ sigma,
                               float* __restrict__ P) {
  int t = threadIdx.x;
  if (t >= 16) return;
  if (t >= 8) {
    for (int m = 0; m < 28; ++m) P[t * 28 + m] = 0.0f;
    return;
  }
  constexpr int PI[28] = {0,0,0,0,0,0,0, 1,1,1,1,1,1, 2,2,2,2,2, 3,3,3,3, 4,4,4, 5,5, 6};
  constexpr int PJ[28] = {0,1,2,3,4,5,6, 1,2,3,4,5,6, 2,3,4,5,6, 3,4,5,6, 4,5,6, 5,6, 6};
  const int k = t;
  float A[3][7];
  for (int o = 0; o < 3; ++o) {
    A[o][0] = W[k * 9 + o * 3 + 0];
    A[o][1] = W[k * 9 + o * 3 + 1];
    A[o][2] = W[k * 9 + o * 3 + 2];
    A[o][3] = (o == 0) ? -1.0f : 0.0f;
    A[o][4] = (o == 1) ? -1.0f : 0.0f;
    A[o][5] = (o == 2) ? -1.0f : 0.0f;
    A[o][6] = bias[k * 3 + o];
  }
  const float s = sigma[0];
  const float scale = -0.5f / (s * s);
  for (int m = 0; m < 28; ++m) {
    const int i = PI[m], j = PJ[m];
    const float q = A[0][i] * A[0][j] + A[1][i] * A[1][j] + A[2][i] * A[2][j];
    P[k * 28 + m] = scale * ((i == j) ? 1.0f : 2.0f) * q;
  }
}

// ---------------------------------------------------------------------------
// Main kernel: one wave32 processes 16 pixels per tile.
// Both 16-lane halves hold the SAME 16 pixels; half 0 supplies K={0,1} of each
// K=4 chunk, half 1 supplies K={2,3} (matches the v_wmma_f32_16x16x4_f32 A/B
// VGPR striping), so no cross-lane shuffles are needed.
// out[b, r, q] for class r comes from accumulator VGPR r, lanes 0..15.
// ---------------------------------------------------------------------------
__global__ void affine_mask_wmma_kernel(const float* __restrict__ img,
                                        const float* __restrict__ simg,
                                        const float* __restrict__ P,
                                        float* __restrict__ out,
                                        int n_tiles) {
  constexpr int PI[28] = {0,0,0,0,0,0,0, 1,1,1,1,1,1, 2,2,2,2,2, 3,3,3,3, 4,4,4, 5,5, 6};
  constexpr int PJ[28] = {0,1,2,3,4,5,6, 1,2,3,4,5,6, 2,3,4,5,6, 3,4,5,6, 4,5,6, 5,6, 6};

  const int lane = threadIdx.x & 31;
  const int hsel = lane >> 4;   // which K-pair of each K=4 chunk this half owns
  const int lp   = lane & 15;   // row (A) / column (B) index within the tile

  const int wave   = blockIdx.x * (blockDim.x >> 5) + (threadIdx.x >> 5);
  const int nwaves = gridDim.x * (blockDim.x >> 5);

  // Preload this lane's A-matrix coefficients (row lp of P, the two K values
  // this half owns in each of the 7 K=4 chunks). 8-byte aligned: lp*28*4 and
  // 2*hsel*4 and 16*j are all multiples of 8.
  v2f a[7];
  const float* prow = P + lp * 28 + 2 * hsel;
#pragma unroll
  for (int j = 0; j < 7; ++j) a[j] = *(const v2f*)(prow + 4 * j);

  for (int t = wave; t < n_tiles; t += nwaves) {
    const int p = t * 16 + lp;          // global pixel id (HW divisible by 16)
    const int b = p >> 18;              // p / (512*512)
    const int q = p & (HW_PIX - 1);     // p % (512*512)

    const float* xb = img  + (b * 3) * HW_PIX + q;
    const float* yb = simg + (b * 3) * HW_PIX + q;
    float z[7];
    z[0] = xb[0]; z[1] = xb[HW_PIX]; z[2] = xb[2 * HW_PIX];
    z[3] = yb[0]; z[4] = yb[HW_PIX]; z[5] = yb[2 * HW_PIX];
    z[6] = 1.0f;

    // u[m] = z_i * z_j over the 28 upper-triangle pairs (registers only).
    float u[28];
#pragma unroll
    for (int m = 0; m < 28; ++m) u[m] = z[PI[m]] * z[PJ[m]];

    v8f acc = {};
#pragma unroll
    for (int j = 0; j < 7; ++j) {
      // B-matrix fragment: this half's two K values for column lp.
      const float b0 = hsel ? u[4 * j + 2] : u[4 * j + 0];
      const float b1 = hsel ? u[4 * j + 3] : u[4 * j + 1];
      v2f bb = {b0, b1};
      acc = __builtin_amdgcn_wmma_f32_16x16x4_f32(
          /*neg_a=*/false, a[j], /*neg_b=*/false, bb,
          /*c_mod=*/(short)0, acc, /*reuse_a=*/false, /*reuse_b=*/false);
    }

    // Rows M=0..7 = classes, valid in lanes 0..15 (lanes 16..31 hold pad rows).
    if (lane < 16) {
      float* ob = out + (b * N_CLS) * HW_PIX + q;
#pragma unroll
      for (int r = 0; r < N_CLS; ++r) ob[r * HW_PIX] = acc[r];
    }
  }
}

// ---------------------------------------------------------------------------
extern "C" void kernel_launch(void* const* d_in, const int* in_sizes, int n_in,
                              void* d_out, int out_size, void* d_ws, size_t ws_size,
                              hipStream_t stream) {
  const float* img   = (const float*)d_in[0];   // [32,3,512,512]
  const float* simg  = (const float*)d_in[1];   // [32,3,512,512]
  const float* W     = (const float*)d_in[2];   // [8,3,3]
  const float* bias  = (const float*)d_in[3];   // [8,3]
  const float* sigma = (const float*)d_in[4];   // [1]
  float* out = (float*)d_out;                   // [32,8,512,512]
  float* P   = (float*)d_ws;                    // 16*28 floats

  setup_P_kernel<<<1, 32, 0, stream>>>(W, bias, sigma, P);

  const int n_tiles = (N_B * HW_PIX) / 16;      // 524288 tiles of 16 pixels
  affine_mask_wmma_kernel<<<2048, 256, 0, stream>>>(img, simg, P, out, n_tiles);
}